// CustomLoss_68049461838137
// MI455X (gfx1250) — compile-verified
//
#include <hip/hip_runtime.h>

#define NN 10
#define CELLS 100
#define BIGV 1000000

typedef __attribute__((ext_vector_type(2))) float v2f;
typedef __attribute__((ext_vector_type(8))) float v8f;

// Computes the reference loss for a single batch element (the last one, since
// the reference returns loss[-1], min_distance[-1]). One 128-thread block.
__global__ __launch_bounds__(128) void CustomLoss_68049461838137_kernel(
    const float* __restrict__ res_g,   // [B,1,10,10]
    const int*   __restrict__ pts_g,   // [B,2,2]
    const float* __restrict__ wm_g,    // [B,1,10,10]
    const float* __restrict__ ww_g,    // [1]
    float* __restrict__ out,           // [2]: loss, min_distance
    int batch) {
  __shared__ float s_res[CELLS];
  __shared__ float s_wm[CELLS];
  __shared__ int   s_lab[CELLS];
  __shared__ int   s_red[2];           // [0]=min pair dist, [1]=start count

  const int tid = threadIdx.x;
  const float* res = res_g + (size_t)batch * CELLS;
  const float* wm  = wm_g  + (size_t)batch * CELLS;

  if (tid == 0) { s_red[0] = BIGV; s_red[1] = 0; }

  bool cellm = false;
  if (tid < CELLS) {
    float v = res[tid];
    s_res[tid] = v;
    s_wm[tid]  = wm[tid];
    cellm = (rintf(v) != 0.0f);           // jnp.round: half-to-even == rintf
    s_lab[tid] = cellm ? (tid + 1) : 0;   // seed labels 1..100 on mask
  }
  __syncthreads();

  // 8-connected min-label propagation, 100 iterations (matches reference scan)
  const int x = tid / NN, y = tid % NN;
  for (int it = 0; it < 100; ++it) {
    int newlab = 0;
    if (tid < CELLS && cellm) {
      int mn = BIGV;
#pragma unroll
      for (int dx = -1; dx <= 1; ++dx) {
#pragma unroll
        for (int dy = -1; dy <= 1; ++dy) {
          int nx = x + dx, ny = y + dy;
          if (nx >= 0 && nx < NN && ny >= 0 && ny < NN) {
            int l = s_lab[nx * NN + ny];      // 0 <=> unmasked <=> BIG in ref
            if (l != 0 && l < mn) mn = l;
          }
        }
      }
      newlab = mn;   // self is masked, so mn < BIG always
    }
    __syncthreads();
    if (tid < CELLS) s_lab[tid] = newlab;
    __syncthreads();
  }

  // Points for this batch
  const int p00 = pts_g[batch * 4 + 0], p01 = pts_g[batch * 4 + 1];
  const int p10 = pts_g[batch * 4 + 2], p11 = pts_g[batch * 4 + 3];
  const int c0 = p00 * NN + p01, c1 = p10 * NN + p11;
  const int sv = s_lab[c0], ev = s_lab[c1];

  // Masked min pairwise L1 distance + start-component size
  bool startm = (tid < CELLS) && (sv > 0) && (s_lab[tid] == sv);
  if (startm) {
    atomicAdd(&s_red[1], 1);
    int lmin = BIGV;
    if (ev > 0) {
      for (int j = 0; j < CELLS; ++j) {
        if (s_lab[j] == ev) {
          int d = abs(x - j / NN) + abs(y - j % NN);
          lmin = min(lmin, d);
        }
      }
    }
    atomicMin(&s_red[0], lmin);
  }
  __syncthreads();

  // sum(res) and sum(res*wm) via V_WMMA_F32_16X16X4_F32.
  // B = all-ones => D[m,n] = sum_k A[m,k]; summing column N=0 over all M gives
  // the sum of every A slot, so the exact (lane,VGPR)->(m,k) bijection is
  // irrelevant. 2 WMMAs cover 128 slots (100 values + zero pad), accumulating
  // in C. Executed uniformly by all 4 waves => EXEC all-1s per wave.
  const int lane = tid & 31;
  const int s0 = lane * 2;

  v2f ones; ones.x = 1.0f; ones.y = 1.0f;
  v8f cr = {0.f, 0.f, 0.f, 0.f, 0.f, 0.f, 0.f, 0.f};
  v8f cw = {0.f, 0.f, 0.f, 0.f, 0.f, 0.f, 0.f, 0.f};

  v2f a;
  a.x = (s0     < CELLS) ? s_res[s0]     : 0.0f;
  a.y = (s0 + 1 < CELLS) ? s_res[s0 + 1] : 0.0f;
  cr = __builtin_amdgcn_wmma_f32_16x16x4_f32(false, a, false, ones, (short)0, cr, false, false);
  a.x = (64 + s0     < CELLS) ? s_res[64 + s0]     : 0.0f;
  a.y = (64 + s0 + 1 < CELLS) ? s_res[64 + s0 + 1] : 0.0f;
  cr = __builtin_amdgcn_wmma_f32_16x16x4_f32(false, a, false, ones, (short)0, cr, false, false);

  v2f aw;
  aw.x = (s0     < CELLS) ? s_res[s0]     * s_wm[s0]     : 0.0f;
  aw.y = (s0 + 1 < CELLS) ? s_res[s0 + 1] * s_wm[s0 + 1] : 0.0f;
  cw = __builtin_amdgcn_wmma_f32_16x16x4_f32(false, aw, false, ones, (short)0, cw, false, false);
  aw.x = (64 + s0     < CELLS) ? s_res[64 + s0]     * s_wm[64 + s0]     : 0.0f;
  aw.y = (64 + s0 + 1 < CELLS) ? s_res[64 + s0 + 1] * s_wm[64 + s0 + 1] : 0.0f;
  cw = __builtin_amdgcn_wmma_f32_16x16x4_f32(false, aw, false, ones, (short)0, cw, false, false);

  // Column N=0 of D: lane 0 holds M=0..7 in c[0..7], lane 16 holds M=8..15.
  float lr = cr[0] + cr[1] + cr[2] + cr[3] + cr[4] + cr[5] + cr[6] + cr[7];
  float lw = cw[0] + cw[1] + cw[2] + cw[3] + cw[4] + cw[5] + cw[6] + cw[7];
  float sum_res = __shfl(lr, 0, 32) + __shfl(lr, 16, 32);
  float sum_rw  = __shfl(lw, 0, 32) + __shfl(lw, 16, 32);

  if (tid == 0) {
    float r0v = s_res[c0];
    float r1v = s_res[c1];
    float manhattan = (float)(abs(p10 - p00) + abs(p11 - p01));

    bool  gap_cond = (sv > 0) && (ev > 0);
    float soa_inv  = 100.0f - sum_res;
    float min_pair = (float)s_red[0];

    float gap_loss = gap_cond ? min_pair * soa_inv * 5000.0f
                              : (2.0f - (r0v + r1v)) * 20000.0f;
    float min_distance = gap_cond ? min_pair : manhattan;

    bool  cond_start = (rintf(r0v) == 0.0f) || (r1v == 0.0f);
    float loss_start = cond_start ? (2.0f - (r0v + r1v)) * 20000.0f : 0.0f;
    float len_start  = gap_cond ? (float)s_red[1] : 0.0f;
    float csp = sum_rw * ww_g[0] * fabsf(manhattan - len_start);

    out[0] = loss_start + csp + gap_loss;
    out[1] = min_distance;
  }
}

extern "C" void kernel_launch(void* const* d_in, const int* in_sizes, int n_in,
                              void* d_out, int out_size, void* d_ws, size_t ws_size,
                              hipStream_t stream) {
  (void)n_in; (void)out_size; (void)d_ws; (void)ws_size;
  const float* res = (const float*)d_in[0];   // result_given [B,1,10,10] f32
  const int*   pts = (const int*)d_in[1];     // points_given [B,2,2] i32
  const float* wm  = (const float*)d_in[2];   // weightmatrix [B,1,10,10] f32
  const float* ww  = (const float*)d_in[3];   // weight_weight [1] f32
  int B = in_sizes[0] / CELLS;
  // Output depends only on the last batch element: O(1) work, one block.
  CustomLoss_68049461838137_kernel<<<1, 128, 0, stream>>>(
      res, pts, wm, ww, (float*)d_out, B - 1);
}